// GraphConvSparse_56556129354624
// MI455X (gfx1250) — compile-verified
//
#include <hip/hip_runtime.h>

// ---------------------------------------------------------------------------
// GraphConv: out = relu( scatter_add(edge_dst, (x@W)[edge_src] * ew) )
//   N=100000 nodes, E=3.2M edges, IN=256, OUT=128, all fp32.
// Phase 1: WMMA f32 GEMM (x @ W) -> xw in d_ws        (memory-bound, ~154MB)
// Phase 2: per-edge gather+atomic scatter into d_out  (L2-resident, dominant)
// Phase 3: in-place ReLU on d_out
// ---------------------------------------------------------------------------

#define IN_DIM  256
#define OUT_DIM 128
#define LDS_STRIDE (IN_DIM + 4)   // +4 dwords: conflict-free ds_load_b64 A-frags

typedef __attribute__((ext_vector_type(2))) float v2f;
typedef __attribute__((ext_vector_type(8))) float v8f;

__global__ __launch_bounds__(256) void gcs_zero_kernel(float4* __restrict__ p, int n4) {
    int i = blockIdx.x * blockDim.x + threadIdx.x;
    if (i < n4) p[i] = make_float4(0.f, 0.f, 0.f, 0.f);
}

// One block = 16 node-rows x all 128 output cols. 8 waves; wave w owns the
// 16x16 column tile starting at w*16. K loop: 64 x V_WMMA_F32_16X16X4_F32.
__global__ __launch_bounds__(256) void gcs_gemm_wmma(const float* __restrict__ x,
                                                     const float* __restrict__ w,
                                                     float* __restrict__ xw,
                                                     int n_nodes) {
    __shared__ float lx[16 * LDS_STRIDE];

    const int tid     = threadIdx.x;
    const int rowbase = blockIdx.x * 16;

    // Cooperatively stage the 16x256 fp32 x-tile (16 KB) into LDS.
    // 1024 float4 loads / 256 threads = 4 each; rows clamped for the tail.
    {
        const float4* x4 = (const float4*)x;            // 64 float4 per row
        #pragma unroll
        for (int i = 0; i < 4; ++i) {
            int v  = tid + i * 256;                     // float4 slot 0..1023
            int r  = v >> 6;                            // row in tile
            int c4 = v & 63;                            // float4 col
            int gr = rowbase + r;
            if (gr >= n_nodes) gr = n_nodes - 1;        // clamp (no OOB read)
            float4 f = x4[(size_t)gr * (IN_DIM / 4) + c4];
            float* dst = &lx[r * LDS_STRIDE + (c4 << 2)];
            dst[0] = f.x; dst[1] = f.y; dst[2] = f.z; dst[3] = f.w;
        }
    }
    __syncthreads();

    const int lane    = tid & 31;
    const int wave    = tid >> 5;          // 0..7 -> 16-col tile
    const int colbase = wave * 16;
    const int mn      = lane & 15;         // A: row m   / B: col n
    const int kk      = (lane >> 4) << 1;  // K sub-offset: 0 or 2

    v8f acc = {};
    const float* wp = w + colbase + mn;    // B[k][n] walk, stride OUT_DIM

    #pragma unroll 8
    for (int k = 0; k < IN_DIM; k += 4) {
        v2f a, b;
        // A frag from LDS: lane holds A[m][k+kk], A[m][k+kk+1]  (ds_load_b64)
        a.x = lx[mn * LDS_STRIDE + k + kk];
        a.y = lx[mn * LDS_STRIDE + k + kk + 1];
        // B frag from global (weight is WGP$/L2 resident, 128 KB total)
        b.x = wp[(size_t)(k + kk) * OUT_DIM];
        b.y = wp[(size_t)(k + kk + 1) * OUT_DIM];
        acc = __builtin_amdgcn_wmma_f32_16x16x4_f32(
                  /*neg_a=*/false, a, /*neg_b=*/false, b,
                  /*c_mod=*/(short)0, acc, /*reuse_a=*/false, /*reuse_b=*/false);
    }

    // D layout: VGPR i -> row (lane>>4)*8 + i, col lane&15
    const int r0 = rowbase + ((lane >> 4) << 3);
    float* o = xw + (size_t)r0 * OUT_DIM + colbase + mn;
    #pragma unroll
    for (int i = 0; i < 8; ++i) {
        if (r0 + i < n_nodes) o[(size_t)i * OUT_DIM] = acc[i];
    }
}

// One wave per edge: edge index is wave-uniform -> scalar loads of
// src/dst/weight; each lane gathers float4 of xw[src] (512B/wave, L2-hit)
// and issues 4 non-returning global_atomic_add_f32 into out[dst].
__global__ __launch_bounds__(256) void gcs_edge_scatter(const float* __restrict__ xw,
                                                        const float* __restrict__ ew,
                                                        const int* __restrict__ esrc,
                                                        const int* __restrict__ edst,
                                                        float* __restrict__ out,
                                                        int n_edges) {
    unsigned t = blockIdx.x * 256u + threadIdx.x;
    unsigned e = t >> 5;
    if (e >= (unsigned)n_edges) return;
    unsigned lane = t & 31u;

    int   s = esrc[e];
    int   d = edst[e];
    float g = ew[e];

    const float4 v = *(const float4*)(xw + (size_t)s * OUT_DIM + lane * 4);
    float* o = out + (size_t)d * OUT_DIM + lane * 4;
    atomicAdd(o + 0, v.x * g);
    atomicAdd(o + 1, v.y * g);
    atomicAdd(o + 2, v.z * g);
    atomicAdd(o + 3, v.w * g);
}

__global__ __launch_bounds__(256) void gcs_relu_kernel(float4* __restrict__ p, int n4) {
    int i = blockIdx.x * blockDim.x + threadIdx.x;
    if (i < n4) {
        float4 v = p[i];
        v.x = fmaxf(v.x, 0.f); v.y = fmaxf(v.y, 0.f);
        v.z = fmaxf(v.z, 0.f); v.w = fmaxf(v.w, 0.f);
        p[i] = v;
    }
}

extern "C" void kernel_launch(void* const* d_in, const int* in_sizes, int n_in,
                              void* d_out, int out_size, void* d_ws, size_t ws_size,
                              hipStream_t stream) {
    const float* x    = (const float*)d_in[0];   // [N, 256]
    const float* w    = (const float*)d_in[1];   // [256, 128]
    const float* ew   = (const float*)d_in[2];   // [E]
    const int*   esrc = (const int*)  d_in[3];   // [E]
    const int*   edst = (const int*)  d_in[4];   // [E]
    float*       out  = (float*)d_out;           // [N, 128]
    float*       xw   = (float*)d_ws;            // [N, 128] scratch (51.2 MB)

    const int n_nodes   = in_sizes[0] / IN_DIM;  // 100000
    const int n_edges   = in_sizes[2];           // 3200000
    const int out_elems = n_nodes * OUT_DIM;     // 12.8M (div by 4)
    const int n4        = out_elems / 4;

    // 1) zero the accumulator (d_out) — required for atomics + determinism
    gcs_zero_kernel<<<(n4 + 255) / 256, 256, 0, stream>>>((float4*)out, n4);

    // 2) xw = x @ W via WMMA f32 16x16x4
    gcs_gemm_wmma<<<(n_nodes + 15) / 16, 256, 0, stream>>>(x, w, xw, n_nodes);

    // 3) gather + weighted atomic scatter-add (one wave per edge)
    unsigned total = (unsigned)n_edges * 32u;
    gcs_edge_scatter<<<(total + 255u) / 256u, 256, 0, stream>>>(xw, ew, esrc, edst,
                                                               out, n_edges);

    // 4) fused activation: in-place ReLU
    gcs_relu_kernel<<<(n4 + 255) / 256, 256, 0, stream>>>((float4*)out, n4);
}